// InstanceLevelFusion_84550726189430
// MI455X (gfx1250) — compile-verified
//
#include <hip/hip_runtime.h>
#include <math.h>

// Polar point-in-box mask: out[p*B + b] = 1.0f if point p falls in box b.
// Store-bandwidth bound: 512 MB of f32 output vs ~1 MB of input.
// Strategy: async-stage the 16KB box table to LDS (CDNA5 async LDS path),
// compute per-box bounds once per block into SoA LDS, then each wave streams
// one output row at a time with 128-bit non-temporal stores.

typedef float v4f __attribute__((ext_vector_type(4)));
typedef int   v4i_b __attribute__((vector_size(16)));   // matches builtin proto

#define TWO_PI_F  6.28318530717958647692f
#define HALF_PI_F 1.57079632679489661923f

// ---- CDNA5 async global->LDS detection (device pass only) ----
#if defined(__has_builtin)
#if __has_builtin(__builtin_amdgcn_global_load_async_to_lds_b128)
#define HAS_ASYNC_LDS 1
#endif
#endif

constexpr int BOXES_MAX     = 1024;          // reference: B = 1024
constexpr int THREADS       = 256;           // 8 waves (wave32)
constexpr int WAVES         = THREADS / 32;
constexpr int PTS_PER_WAVE  = 4;
constexpr int PTS_PER_BLOCK = WAVES * PTS_PER_WAVE;  // 32

__global__ __launch_bounds__(THREADS)
void polar_mask_kernel(const float* __restrict__ pts,      // (P,2)
                       const float* __restrict__ corners,  // (B,4)
                       float* __restrict__ out,            // (P,B) as f32
                       int P, int B) {
    __shared__ v4f  s_raw[BOXES_MAX];                       // raw corners, 16 KB
    __shared__ float s_rlow[BOXES_MAX];                     // SoA bounds, 16 KB
    __shared__ float s_rup [BOXES_MAX];
    __shared__ float s_amin[BOXES_MAX];
    __shared__ float s_amax[BOXES_MAX];

    if (B > BOXES_MAX) return;   // layout assumption (reference: B == 1024)

    // ---- Stage corner table into LDS ----
#if HAS_ASYNC_LDS
    {
        for (int i = threadIdx.x; i < B; i += THREADS) {
            // 16 bytes (one box's 4 corners) per lane, memory -> LDS, no VGPR data.
            __builtin_amdgcn_global_load_async_to_lds_b128(
                (__attribute__((address_space(1))) v4i_b*)
                    ((const char*)corners + (size_t)i * 16),
                (__attribute__((address_space(3))) v4i_b*)
                    ((char*)s_raw + i * 16),
                0, 0);
        }
#if __has_builtin(__builtin_amdgcn_s_wait_asynccnt)
        __builtin_amdgcn_s_wait_asynccnt(0);
#else
        asm volatile("s_wait_asynccnt 0" ::: "memory");
#endif
    }
#else
    for (int i = threadIdx.x; i < B; i += THREADS)
        s_raw[i] = ((const v4f*)corners)[i];
#endif
    __syncthreads();

    // ---- Per-box bounds (exactly mirrors the reference math) ----
    for (int b = threadIdx.x; b < B; b += THREADS) {
        v4f c = s_raw[b];
        float r_max = c.x, r_min = c.y;
        float range_center = 0.5f * (r_max + r_min);
        float box_length   = fabsf(r_max - r_min);
        float range_off    = fminf(box_length * 1.05f, 5.0f);
        s_rlow[b] = r_min - range_off;
        s_rup [b] = r_max + range_off;

        float a_max = (c.z < 0.0f) ? c.z + TWO_PI_F : c.z;
        float a_min = (c.w < 0.0f) ? c.w + TWO_PI_F : c.w;
        float abs_azi = fabsf(a_max - a_min);
        abs_azi = (abs_azi > HALF_PI_F) ? TWO_PI_F - abs_azi : abs_azi;
        float box_width = abs_azi * range_center;
        float width_off = fminf(box_width * 1.05f, 5.0f);
        float azi_off   = fminf(fmaxf(width_off / range_center, 0.0f), 0.05f);
        float amax = a_max + azi_off;
        float amin = a_min - azi_off;
        s_amax[b] = (amax > TWO_PI_F) ? amax - TWO_PI_F : amax;
        s_amin[b] = (amin < 0.0f)    ? amin + TWO_PI_F : amin;
    }
    __syncthreads();

    // ---- Stream output rows: one wave per point, 4 boxes per lane ----
    const int wave = threadIdx.x >> 5;
    const int lane = threadIdx.x & 31;
    const v4f* rl4 = (const v4f*)s_rlow;
    const v4f* ru4 = (const v4f*)s_rup;
    const v4f* an4 = (const v4f*)s_amin;
    const v4f* ax4 = (const v4f*)s_amax;
    const int ngroups = B >> 2;   // float4 groups per row

    for (int j = 0; j < PTS_PER_WAVE; ++j) {
        int p = blockIdx.x * PTS_PER_BLOCK + wave * PTS_PER_WAVE + j;
        if (p >= P) break;
        float pr  = pts[2 * p + 0];
        float paw = pts[2 * p + 1];
        float pa  = (paw < 0.0f) ? paw + TWO_PI_F : paw;

        v4f* outrow = (v4f*)(out + (size_t)p * (size_t)B);
        for (int g = lane; g < ngroups; g += 32) {
            v4f rl = rl4[g], ru = ru4[g], an = an4[g], ax = ax4[g];
            v4f m;
            m.x = (pr < ru.x && pr > rl.x && pa > an.x && pa < ax.x) ? 1.0f : 0.0f;
            m.y = (pr < ru.y && pr > rl.y && pa > an.y && pa < ax.y) ? 1.0f : 0.0f;
            m.z = (pr < ru.z && pr > rl.z && pa > an.z && pa < ax.z) ? 1.0f : 0.0f;
            m.w = (pr < ru.w && pr > rl.w && pa > an.w && pa < ax.w) ? 1.0f : 0.0f;
            // 512 MB output stream > 192 MB L2: bypass with non-temporal b128 store.
            __builtin_nontemporal_store(m, outrow + g);
        }
    }
}

extern "C" void kernel_launch(void* const* d_in, const int* in_sizes, int n_in,
                              void* d_out, int out_size, void* d_ws, size_t ws_size,
                              hipStream_t stream) {
    const float* pts     = (const float*)d_in[0];   // polar_pts (P,2) f32
    const float* corners = (const float*)d_in[1];   // polar_corners (B,4) f32
    float* out = (float*)d_out;                     // (P,B) mask as f32

    const int P = in_sizes[0] / 2;
    const int B = in_sizes[1] / 4;

    const int blocks = (P + PTS_PER_BLOCK - 1) / PTS_PER_BLOCK;  // 4096 for P=131072
    polar_mask_kernel<<<blocks, THREADS, 0, stream>>>(pts, corners, out, P, B);
}